// Sampler_28449863369149
// MI455X (gfx1250) — compile-verified
//
#include <hip/hip_runtime.h>
#include <math.h>

#define N_ROWS   256
#define VOCAB    128000
#define P_LEN    512
#define O_LEN    128
#define NLOGPROB 20
#define NTHREADS 256
#define NWAVES   (NTHREADS / 32)
#define WCHUNK   128                    // floats per wave per chunk (1 b128 line per lane)
#define CHUNK    (NTHREADS * 4)         // 1024 floats per block chunk
#define NCHUNKS  (VOCAB / CHUNK)        // 125, exact
#define NBUF     8                      // async pipeline depth (power of two)
#define NBINS    2048
#define NGROUPS  (NBINS / 16)
#define CAND_PEN_CAP 512
#define CAND_RAW_CAP 256
#define MASK_WORDS   ((VOCAB + 31) / 32)  // 4000

// ---------- order-preserving float <-> uint key ----------
__device__ __forceinline__ unsigned f2key(float f) {
  unsigned u = __float_as_uint(f);
  return (u & 0x80000000u) ? ~u : (u | 0x80000000u);
}
__device__ __forceinline__ float key2f(unsigned k) {
  unsigned u = (k & 0x80000000u) ? (k & 0x7FFFFFFFu) : ~k;
  return __uint_as_float(u);
}

// ---------- CDNA5 async global->LDS streaming ----------
__device__ __forceinline__ unsigned lds_off(const void* p) {
  // generic LDS pointer: low 32 bits are the LDS byte offset (ISA 10.2 aperture rule)
  return (unsigned)(unsigned long long)p;
}
__device__ __forceinline__ void async_b128(unsigned lds, const void* gaddr) {
  asm volatile("global_load_async_to_lds_b128 %0, %1, off"
               :: "v"(lds), "v"(gaddr) : "memory");
}
__device__ __forceinline__ void wait_async_le(int n) {
  switch (n) {
    case 0:  asm volatile("s_wait_asynccnt 0x0" ::: "memory"); break;
    case 1:  asm volatile("s_wait_asynccnt 0x1" ::: "memory"); break;
    case 2:  asm volatile("s_wait_asynccnt 0x2" ::: "memory"); break;
    case 3:  asm volatile("s_wait_asynccnt 0x3" ::: "memory"); break;
    case 4:  asm volatile("s_wait_asynccnt 0x4" ::: "memory"); break;
    case 5:  asm volatile("s_wait_asynccnt 0x5" ::: "memory"); break;
    case 6:  asm volatile("s_wait_asynccnt 0x6" ::: "memory"); break;
    case 7:  asm volatile("s_wait_asynccnt 0x7" ::: "memory"); break;
    case 8:  asm volatile("s_wait_asynccnt 0x8" ::: "memory"); break;
    case 9:  asm volatile("s_wait_asynccnt 0x9" ::: "memory"); break;
    case 10: asm volatile("s_wait_asynccnt 0xa" ::: "memory"); break;
    case 11: asm volatile("s_wait_asynccnt 0xb" ::: "memory"); break;
    case 12: asm volatile("s_wait_asynccnt 0xc" ::: "memory"); break;
    case 13: asm volatile("s_wait_asynccnt 0xd" ::: "memory"); break;
    default: asm volatile("s_wait_asynccnt 0xe" ::: "memory"); break;
  }
}

// Stream one row (logits [+ mask]) through per-wave LDS multi-buffers.
// Each wave loads and consumes only its own LDS slice -> no block barriers inside.
// Async loads complete in order per wave, so waiting ASYNCcnt <= ahead*I
// guarantees chunk c's data has landed in LDS.
template <bool WITH_MASK, typename F>
__device__ void stream_pass(const float* __restrict__ lrow, const int* __restrict__ mrow,
                            float (&lbuf)[NWAVES][NBUF][WCHUNK],
                            int (&mbuf)[NWAVES][NBUF][WCHUNK], F&& fn) {
  const int w = threadIdx.x >> 5, lane = threadIdx.x & 31;
  const int off = w * WCHUNK + lane * 4;
  const int I = WITH_MASK ? 2 : 1;
#pragma unroll
  for (int c = 0; c < NBUF - 1; ++c) {
    if (c < NCHUNKS) {
      async_b128(lds_off(&lbuf[w][c][lane * 4]), lrow + c * CHUNK + off);
      if (WITH_MASK) async_b128(lds_off(&mbuf[w][c][lane * 4]), mrow + c * CHUNK + off);
    }
  }
  for (int c = 0; c < NCHUNKS; ++c) {
    const int ia = c + NBUF - 1;
    if (ia < NCHUNKS) {
      const int b = ia & (NBUF - 1);
      async_b128(lds_off(&lbuf[w][b][lane * 4]), lrow + ia * CHUNK + off);
      if (WITH_MASK) async_b128(lds_off(&mbuf[w][b][lane * 4]), mrow + ia * CHUNK + off);
    }
    int ahead = NCHUNKS - 1 - c; if (ahead > NBUF - 1) ahead = NBUF - 1;
    wait_async_le(ahead * I);
    const int cb = c & (NBUF - 1);
    const float4 xs = *(const float4*)&lbuf[w][cb][lane * 4];
    int4 ms; ms.x = ms.y = ms.z = ms.w = 0;
    if (WITH_MASK) ms = *(const int4*)&mbuf[w][cb][lane * 4];
    const int vb = c * CHUNK + off;
    fn(vb + 0, xs.x, ms.x); fn(vb + 1, xs.y, ms.y);
    fn(vb + 2, xs.z, ms.z); fn(vb + 3, xs.w, ms.w);
  }
}

// ---------- block reductions (deterministic trees) ----------
__device__ unsigned long long reduce_max_u64(unsigned long long v, unsigned long long* buf) {
  const int t = threadIdx.x;
  buf[t] = v; __syncthreads();
  for (int s = NTHREADS / 2; s > 0; s >>= 1) {
    if (t < s) { unsigned long long o = buf[t + s]; if (o > buf[t]) buf[t] = o; }
    __syncthreads();
  }
  unsigned long long r = buf[0]; __syncthreads();
  return r;
}
__device__ unsigned long long reduce_add_u64(unsigned long long v, unsigned long long* buf) {
  const int t = threadIdx.x;
  buf[t] = v; __syncthreads();
  for (int s = NTHREADS / 2; s > 0; s >>= 1) {
    if (t < s) buf[t] += buf[t + s];
    __syncthreads();
  }
  unsigned long long r = buf[0]; __syncthreads();
  return r;
}
__device__ double reduce_add_d(double v, double* buf) {
  const int t = threadIdx.x;
  buf[t] = v; __syncthreads();
  for (int s = NTHREADS / 2; s > 0; s >>= 1) {
    if (t < s) buf[t] += buf[t + s];
    __syncthreads();
  }
  double r = buf[0]; __syncthreads();
  return r;
}

// descending bitonic sort of n (power of two) u64 keys in LDS
__device__ void bitonic_sort_desc(unsigned long long* a, int n) {
  for (int k = 2; k <= n; k <<= 1)
    for (int j = k >> 1; j > 0; j >>= 1) {
      for (int i = threadIdx.x; i < n; i += NTHREADS) {
        const int ixj = i ^ j;
        if (ixj > i) {
          unsigned long long x = a[i], y = a[ixj];
          const bool sw = ((i & k) == 0) ? (x < y) : (x > y);
          if (sw) { a[i] = y; a[ixj] = x; }
        }
      }
      __syncthreads();
    }
}

// find largest bin b such that sum(hist[b..NBINS-1]) >= K
__device__ int scan_cut(const unsigned* gs, const unsigned* hist, unsigned K) {
  unsigned acc = 0;
  for (int g = NGROUPS - 1; g >= 0; --g) {
    if (acc + gs[g] >= K) {
      for (int b = g * 16 + 15;; --b) {
        acc += hist[b];
        if (acc >= K || b == g * 16) return b;
      }
    }
    acc += gs[g];
  }
  return 0;
}

__global__ __launch_bounds__(NTHREADS) void sampler_kernel(
    const float* __restrict__ logits, const int* __restrict__ allowed_mask,
    const int* __restrict__ prompt_ids, const int* __restrict__ output_ids,
    const float* __restrict__ presence, const float* __restrict__ frequency,
    const float* __restrict__ repetition, const float* __restrict__ temperature,
    const int* __restrict__ top_k, const float* __restrict__ top_p,
    const float* __restrict__ noise, float* __restrict__ out) {
  const int row = blockIdx.x;
  if (row >= N_ROWS) return;
  const int tid = threadIdx.x;

  __shared__ unsigned seen_bits[MASK_WORDS];
  __shared__ unsigned out_bits[MASK_WORDS];
  __shared__ int out_toks[O_LEN];
  __shared__ unsigned hist_pen[NBINS];
  __shared__ unsigned hist_raw[NBINS];
  __shared__ __align__(16) float lbuf[NWAVES][NBUF][WCHUNK];
  __shared__ __align__(16) int   mbuf[NWAVES][NBUF][WCHUNK];
  __shared__ unsigned long long cand_pen[CAND_PEN_CAP];
  __shared__ unsigned long long cand_raw[CAND_RAW_CAP];
  __shared__ unsigned long long r_u64[NTHREADS];
  __shared__ double r_d[NTHREADS];
  __shared__ unsigned gsA[NGROUPS], gsB[NGROUPS];
  __shared__ float sh_p[64], sh_sc[64];
  __shared__ unsigned sh_idx[64];
  __shared__ unsigned cnt_pen, cnt_raw;
  __shared__ int sh_cut_pen, sh_cut_raw, sh_sampled;
  __shared__ float sh_toklp;

  // ---------- phase 0: init masks / tables ----------
  for (int i = tid; i < MASK_WORDS; i += NTHREADS) { seen_bits[i] = 0; out_bits[i] = 0; }
  for (int i = tid; i < NBINS; i += NTHREADS) { hist_pen[i] = 0; hist_raw[i] = 0; }
  for (int i = tid; i < CAND_PEN_CAP; i += NTHREADS) cand_pen[i] = 0;
  for (int i = tid; i < CAND_RAW_CAP; i += NTHREADS) cand_raw[i] = 0;
  if (tid == 0) { cnt_pen = 0; cnt_raw = 0; }
  if (tid < O_LEN) {
    const int t = output_ids[row * O_LEN + tid];
    out_toks[tid] = t;
    atomicOr(&out_bits[t >> 5], 1u << (t & 31));
    atomicOr(&seen_bits[t >> 5], 1u << (t & 31));
  }
  for (int i = tid; i < P_LEN; i += NTHREADS) {
    const int t = prompt_ids[row * P_LEN + i];
    atomicOr(&seen_bits[t >> 5], 1u << (t & 31));
  }
  __syncthreads();

  const float rep = repetition[row], fr = frequency[row], pr = presence[row];
  const float temp = temperature[row];
  const float stemp = (temp < 1e-5f) ? 1.0f : temp;
  const float inv_t = 1.0f / stemp;
  const int tk = top_k[row];
  const float tp = top_p[row];
  const float NEG_INF = -__builtin_huge_valf();
  const float* lrow = logits + (size_t)row * VOCAB;
  const int* mrow = allowed_mask + (size_t)row * VOCAB;

  auto penalize = [&](int v, float x, int m) -> float {
    float xp = m ? NEG_INF : x;                       // allowed_mask -> -inf
    const bool sn = (seen_bits[v >> 5] >> (v & 31)) & 1;
    if (sn) xp = (xp > 0.f) ? (xp / rep) : (xp * rep);
    const bool ob = (out_bits[v >> 5] >> (v & 31)) & 1;
    if (ob) {
      float cnt = 0.f;
      for (int j = 0; j < O_LEN; ++j) cnt += (out_toks[j] == v) ? 1.f : 0.f;
      xp = xp - fr * cnt;
      xp = xp - pr;
    }
    return xp;
  };

  // ---------- phase 1: maxes, greedy argmax, histograms ----------
  unsigned long long gmax = 0ull, rmax = 0ull;
  stream_pass<true>(lrow, mrow, lbuf, mbuf, [&](int v, float x, int m) {
    const float xp = penalize(v, x, m);
    const unsigned kp = f2key(xp), kr = f2key(x);
    atomicAdd(&hist_pen[kp >> 21], 1u);
    atomicAdd(&hist_raw[kr >> 21], 1u);
    const unsigned long long pp = ((unsigned long long)kp << 32) | (unsigned)(~v);
    const unsigned long long rr = ((unsigned long long)kr << 32) | (unsigned)(~v);
    if (pp > gmax) gmax = pp;
    if (rr > rmax) rmax = rr;
  });
  __syncthreads();
  gmax = reduce_max_u64(gmax, r_u64);
  rmax = reduce_max_u64(rmax, r_u64);
  const float m_pen = key2f((unsigned)(gmax >> 32));
  const float m_raw = key2f((unsigned)(rmax >> 32));
  const int greedy_idx = (int)(~(unsigned)gmax);

  if (tid < NGROUPS) {
    unsigned a = 0, b = 0;
    for (int j = 0; j < 16; ++j) { a += hist_pen[tid * 16 + j]; b += hist_raw[tid * 16 + j]; }
    gsA[tid] = a; gsB[tid] = b;
  }
  __syncthreads();
  if (tid == 0)  sh_cut_pen = scan_cut(gsA, hist_pen, 64u);
  if (tid == 32) sh_cut_raw = scan_cut(gsB, hist_raw, (unsigned)NLOGPROB);
  __syncthreads();
  const int cutp = sh_cut_pen, cutr = sh_cut_raw;

  // ---------- phase 2: exp-sums + candidate collection ----------
  double sp = 0.0, sr = 0.0;
  stream_pass<true>(lrow, mrow, lbuf, mbuf, [&](int v, float x, int m) {
    const float xp = penalize(v, x, m);
    const unsigned kp = f2key(xp), kr = f2key(x);
    sr += (double)expf(x - m_raw);
    sp += (double)expf((xp - m_pen) * inv_t);
    if ((int)(kp >> 21) >= cutp) {
      const unsigned pos = atomicAdd(&cnt_pen, 1u);
      if (pos < CAND_PEN_CAP)
        cand_pen[pos] = ((unsigned long long)kp << 32) | (unsigned)(~v);
    }
    if ((int)(kr >> 21) >= cutr) {
      const unsigned pos = atomicAdd(&cnt_raw, 1u);
      if (pos < CAND_RAW_CAP)
        cand_raw[pos] = ((unsigned long long)kr << 32) | (unsigned)(~v);
    }
  });
  __syncthreads();
  const double sum_raw = reduce_add_d(sr, r_d);
  const double sum_pen = reduce_add_d(sp, r_d);
  const float lse_log = logf((float)sum_raw);

  bitonic_sort_desc(cand_pen, CAND_PEN_CAP);
  bitonic_sort_desc(cand_raw, CAND_RAW_CAP);

  // ---------- sampling over top-64 candidates ----------
  int M = (int)(cnt_pen < (unsigned)CAND_PEN_CAP ? cnt_pen : (unsigned)CAND_PEN_CAP);
  if (M > 64) M = 64;
  const float mt = m_pen / stemp;
  const float sumpen_f = (float)sum_pen;
  if (tid < M) {
    const unsigned long long e = cand_pen[tid];
    const unsigned key = (unsigned)(e >> 32);
    const unsigned idx = ~(unsigned)e;
    const float srt = key2f(key) / stemp;               // match reference rounding
    sh_p[tid] = expf(srt - mt) / sumpen_f;
    const float u = noise[(size_t)row * VOCAB + idx];
    const float g = -logf(-logf(u + 1e-20f) + 1e-20f);
    sh_sc[tid] = srt + g;
    sh_idx[tid] = idx;
  }
  __syncthreads();
  if (tid == 0) {
    float cum = 0.f, bs = 0.f;
    unsigned bidx = 0; bool have = false;
    for (int i = 0; i < M; ++i) {
      const float p = sh_p[i];
      const bool masked = (i >= tk) || (cum > tp);      // cum here == cum_i - p_i
      cum += p;
      if (!masked) {
        const float sc = sh_sc[i];
        if (!have || sc > bs || (sc == bs && sh_idx[i] < bidx)) {
          have = true; bs = sc; bidx = sh_idx[i];
        }
      }
    }
    const int sampled = (temp < 1e-5f) ? greedy_idx : (int)bidx;
    sh_sampled = sampled;
    const float xs = lrow[sampled];
    const float toklp = (xs - m_raw) - lse_log;
    sh_toklp = toklp;
    out[row] = (float)sampled;                                   // sampled
    out[N_ROWS + row * 21 + 0] = (float)sampled;                 // out_indices[:,0]
    out[N_ROWS * 22 + row * 21 + 0] = toklp;                     // out_logprobs[:,0]
  }
  // top-20 logprobs
  const int MR = (int)(cnt_raw < (unsigned)CAND_RAW_CAP ? cnt_raw : (unsigned)CAND_RAW_CAP);
  if (tid < NLOGPROB) {
    float lp; unsigned idx;
    if (tid < MR) {
      const unsigned long long e = cand_raw[tid];
      idx = ~(unsigned)e;
      lp = (key2f((unsigned)(e >> 32)) - m_raw) - lse_log;
    } else { idx = 0; lp = NEG_INF; }
    out[N_ROWS + row * 21 + 1 + tid] = (float)idx;
    out[N_ROWS * 22 + row * 21 + 1 + tid] = lp;
  }
  __syncthreads();

  // ---------- phase 3: token rank ----------
  const float toklp = sh_toklp;
  unsigned long long rk = 0ull;
  stream_pass<false>(lrow, mrow, lbuf, mbuf, [&](int v, float x, int m) {
    (void)v; (void)m;
    const float rl = (x - m_raw) - lse_log;
    rk += (rl >= toklp) ? 1ull : 0ull;
  });
  __syncthreads();
  rk = reduce_add_u64(rk, r_u64);
  if (tid == 0) out[N_ROWS * 43 + row] = (float)rk;
  (void)sh_sampled;
}

extern "C" void kernel_launch(void* const* d_in, const int* in_sizes, int n_in,
                              void* d_out, int out_size, void* d_ws, size_t ws_size,
                              hipStream_t stream) {
  (void)in_sizes; (void)n_in; (void)out_size; (void)d_ws; (void)ws_size;
  const float* logits  = (const float*)d_in[0];
  const int*   allowed = (const int*)d_in[1];
  const int*   prompt  = (const int*)d_in[2];
  const int*   outids  = (const int*)d_in[3];
  const float* pres    = (const float*)d_in[4];
  const float* freq    = (const float*)d_in[5];
  const float* rep     = (const float*)d_in[6];
  const float* temp    = (const float*)d_in[7];
  const int*   topk    = (const int*)d_in[8];
  const float* topp    = (const float*)d_in[9];
  const float* noise   = (const float*)d_in[10];
  sampler_kernel<<<N_ROWS, NTHREADS, 0, stream>>>(
      logits, allowed, prompt, outids, pres, freq, rep, temp, topk, topp, noise,
      (float*)d_out);
}